// DilationLayerExtSE_83193516523621
// MI455X (gfx1250) — compile-verified
//
#include <hip/hip_runtime.h>
#include <hip/hip_bf16.h>

// Morphological dilation 5x5, pad=2, stride=1, per-(b,c) weights.
// B=8, C=128, H=W=128. One workgroup per half-plane (64 output rows).
//
// LDS tile: 68 rows x 136-float stride.
//   T[tr][c] == xp[i0g + tr][c - 2]   (xp = zero-padded plane, 132x132)
//   interior x[rx][col] stored at T[tr][col + 4]  -> 16B-aligned b128 copies
//   compute reads cols 2..133, rows 0..67 only.

#define TSTRIDE   136
#define TROWS     68
#define TFLOATS   (TROWS * TSTRIDE)          // 9248 floats = 36992 bytes
#define PLANE_HW  (128 * 128)

#if __has_builtin(__builtin_amdgcn_global_load_async_to_lds_b128) && \
    __has_builtin(__builtin_amdgcn_s_wait_asynccnt)
#define USE_ASYNC_LDS 1
typedef int v4i __attribute__((vector_size(16)));
#else
#define USE_ASYNC_LDS 0
#endif

__global__ __launch_bounds__(256)
void dilation_max5x5_kernel(const float* __restrict__ x,
                            const float* __restrict__ weight,
                            const float* __restrict__ bias,
                            float* __restrict__ out)
{
    __shared__ float T[TFLOATS];

    const int tid = threadIdx.x;
    const int p   = blockIdx.x >> 1;     // plane index: b*128 + c
    const int bi  = blockIdx.x & 1;      // half-plane (output rows bi*64 .. bi*64+63)
    const int i0g = bi * 64;

    const float* __restrict__ xp = x   + (size_t)p * PLANE_HW;
    float*       __restrict__ op = out + (size_t)p * PLANE_HW;

    // ---- 1. zero the whole tile (covers the zero-pad halo) -----------------
    {
        float4* T4 = (float4*)T;
        const float4 z = make_float4(0.f, 0.f, 0.f, 0.f);
        for (int e = tid; e < TFLOATS / 4; e += 256)
            T4[e] = z;
    }
    __syncthreads();

    // ---- 2. stage interior rows: x rows [rx_start, rx_start+65] ------------
    // tile row tr corresponds to xp row i0g+tr; x row rx = i0g+tr-2.
    const int rx_start = (i0g - 2 < 0) ? 0 : (i0g - 2);
    const int tr_start = rx_start - (i0g - 2);   // 2 for bi==0, 0 for bi==1
    // 66 rows x 32 float4 each = 2112 transfers
    for (int e = tid; e < 66 * 32; e += 256) {
        const int rl = e >> 5;          // 0..65
        const int c4 = e & 31;          // 0..31
        const float* g = xp + (rx_start + rl) * 128 + c4 * 4;
        float*       l = T  + (tr_start + rl) * TSTRIDE + 4 + c4 * 4;
#if USE_ASYNC_LDS
        __builtin_amdgcn_global_load_async_to_lds_b128(
            (v4i*)g, (v4i*)l, 0, 0);
#else
        *(float4*)l = *(const float4*)g;
#endif
    }
#if USE_ASYNC_LDS
    __builtin_amdgcn_s_wait_asynccnt(0);
#endif
    __syncthreads();

    // ---- 3. per-plane weights / bias (block-uniform -> scalar loads) -------
    const float* __restrict__ wp = weight + p * 25;
    float w[25];
#pragma unroll
    for (int k = 0; k < 25; ++k) w[k] = wp[k];
    const float bv = bias[p];

    // ---- 4. compute: each thread does two 4x4 output blocks ----------------
#pragma unroll 1
    for (int t = 0; t < 2; ++t) {
        const int blk = tid + t * 256;        // 0..511
        const int il  = (blk >> 5) * 4;       // local output row base 0..60
        const int j0  = (blk & 31) * 4;       // output col base 0..124

        float acc[4][4];
#pragma unroll
        for (int r = 0; r < 4; ++r)
#pragma unroll
            for (int c = 0; c < 4; ++c)
                acc[r][c] = -__builtin_huge_valf();

        // stream 8 tile rows; row (il+r8) contributes to outputs oi = r8-di
#pragma unroll
        for (int r8 = 0; r8 < 8; ++r8) {
            float rb[8];
            const float* rp = T + (il + r8) * TSTRIDE + j0 + 2;
#pragma unroll
            for (int k = 0; k < 8; ++k) rb[k] = rp[k];

#pragma unroll
            for (int di = 0; di < 5; ++di) {
                const int oi = r8 - di;             // compile-time resolved
                if (oi >= 0 && oi < 4) {
#pragma unroll
                    for (int dj = 0; dj < 5; ++dj) {
                        const float wv = w[di * 5 + dj];
#pragma unroll
                        for (int jj = 0; jj < 4; ++jj)
                            acc[oi][jj] = fmaxf(acc[oi][jj], rb[dj + jj] + wv);
                    }
                }
            }
        }

        // coalesced 16B stores
#pragma unroll
        for (int r = 0; r < 4; ++r) {
            float4 v = make_float4(acc[r][0] + bv, acc[r][1] + bv,
                                   acc[r][2] + bv, acc[r][3] + bv);
            *(float4*)(op + (i0g + il + r) * 128 + j0) = v;
        }
    }
}

extern "C" void kernel_launch(void* const* d_in, const int* in_sizes, int n_in,
                              void* d_out, int out_size, void* d_ws, size_t ws_size,
                              hipStream_t stream) {
    const float* x      = (const float*)d_in[0];
    const float* weight = (const float*)d_in[1];
    const float* bias   = (const float*)d_in[2];
    // d_in[3] = padding (2), d_in[4] = stride (1): fixed by setup_inputs.
    float* out = (float*)d_out;

    dim3 grid(8 * 128 * 2);   // 2048 half-plane workgroups
    dim3 block(256);
    hipLaunchKernelGGL(dilation_max5x5_kernel, grid, block, 0, stream,
                       x, weight, bias, out);
}